// Gaussians_90151363543778
// MI455X (gfx1250) — compile-verified
//
#include <hip/hip_runtime.h>
#include <math.h>

typedef float v2f __attribute__((ext_vector_type(2)));
typedef float v8f __attribute__((ext_vector_type(8)));

#define N_PTS 12288
#define BASE_SCALE 0.001f
#define INF_BITS 0x7f800000u

__device__ __forceinline__ unsigned umn(unsigned a, unsigned b) { return a < b ? a : b; }
__device__ __forceinline__ unsigned umx(unsigned a, unsigned b) { return a > b ? a : b; }

// Insert v into ascending sorted triple (x0 <= x1 <= x2) keeping 3 smallest.
// Operates on bit patterns of non-negative floats: unsigned compare == float
// compare for values >= 0, and v_min_u32/v_max_u32 need no canonicalization.
__device__ __forceinline__ void ins3u(unsigned v, unsigned& x0, unsigned& x1, unsigned& x2) {
    unsigned lo0 = umn(v, x0);
    unsigned hi0 = umx(v, x0);
    x0 = lo0;
    unsigned lo1 = umn(hi0, x1);
    unsigned hi1 = umx(hi0, x1);
    x1 = lo1;
    x2 = umn(hi1, x2);
}

// One 4-wave workgroup per 16-row tile. Each wave scans a strided quarter of
// the column tiles with V_WMMA_F32_16X16X4_F32; partial top-3 sets are merged
// via shuffles (within-wave) then LDS (across waves). Scale s for point i is
// stashed in out9[i*9].
__global__ __launch_bounds__(128) void knn_scale_kernel(const float* __restrict__ pts,
                                                        float* __restrict__ out9) {
    __shared__ unsigned red[4][16][3];   // [wave][row][top3 bits]

    const int tid  = threadIdx.x;
    // Force the wave index into an SGPR so the diagonal test below becomes a
    // true scalar branch (s_cmp + s_cbranch), not an EXEC-mask divergent one.
    const int wave = __builtin_amdgcn_readfirstlane(tid >> 5);
    const int lane = tid & 31;
    const int half = lane >> 4;   // 0 or 1
    const int l    = lane & 15;
    const int rowBase = blockIdx.x * 16;

    // A operand (16x4 f32): lanes 0-15 -> K=0,1 ; lanes 16-31 -> K=2,3.
    // Rows pre-scaled by -2; K=3 column set to 1.0 so that, with
    // B[3][N] = |P_col|^2, the WMMA tile is (-2*dot + sqCol).
    const float* prow = pts + (size_t)(rowBase + l) * 3;
    float rx = prow[0], ry = prow[1], rz = prow[2];
    v2f a;
    a.x = (half == 0) ? (-2.0f * rx) : (-2.0f * rz);
    a.y = (half == 0) ? (-2.0f * ry) : 1.0f;

    // |P|^2 for the 8 rows this lane accumulates in the D tile (M = r + 8*half)
    float sqRow[8];
    #pragma unroll
    for (int r = 0; r < 8; ++r) {
        const float* p = pts + (size_t)(rowBase + r + 8 * half) * 3;
        float x = p[0], y = p[1], z = p[2];
        sqRow[r] = x * x + y * y + z * z;
    }

    unsigned t0[8], t1[8], t2[8];
    #pragma unroll
    for (int r = 0; r < 8; ++r) { t0[r] = INF_BITS; t1[r] = INF_BITS; t2[r] = INF_BITS; }

    // Compile-time trip count (192, provably even) -> clean unroll-by-2,
    // no guard/peel blocks, two b96 loads in flight per iteration.
    #pragma unroll 2
    for (int it = 0; it < N_PTS / 64; ++it) {
        const int c = it * 64 + wave * 16;   // scalar column-tile base

        // B operand (4x16 f32): VGPR0 -> K = 2*half, VGPR1 -> K = 2*half+1, col N = l
        const float* pcol = pts + (size_t)(c + l) * 3;
        float cx = pcol[0], cy = pcol[1], cz = pcol[2];
        float sqCol = cx * cx + cy * cy + cz * cz;
        v2f b;
        b.x = (half == 0) ? cx : cz;
        b.y = (half == 0) ? cy : sqCol;

        v8f acc = {0.f, 0.f, 0.f, 0.f, 0.f, 0.f, 0.f, 0.f};
        // acc[r] = -2 * dot(P[rowBase + r + 8*half], P[c + l]) + |P[c + l]|^2
        acc = __builtin_amdgcn_wmma_f32_16x16x4_f32(
            /*neg_a=*/false, a, /*neg_b=*/false, b,
            /*c_mod=*/(short)0, acc, /*reuse_a=*/false, /*reuse_b=*/false);

        if (c != rowBase) {   // scalar (SGPR) condition -> s_cbranch, EXEC untouched
            #pragma unroll
            for (int r = 0; r < 8; ++r) {
                float d2 = fmaxf(sqRow[r] + acc[r], 0.0f);   // >= 0, non-NaN
                ins3u(__float_as_uint(d2), t0[r], t1[r], t2[r]);
            }
        } else {
            const int colIdx = c + l;
            #pragma unroll
            for (int r = 0; r < 8; ++r) {
                float d2 = fmaxf(sqRow[r] + acc[r], 0.0f);
                unsigned u = __float_as_uint(d2);
                const int rowIdx = rowBase + r + 8 * half;
                if (colIdx == rowIdx) u = INF_BITS;   // exclude diagonal (cndmask)
                ins3u(u, t0[r], t1[r], t2[r]);
            }
        }
    }

    // Butterfly merge of the per-lane top-3 across the 16 lanes of the same half.
    // xor masks 1,2,4,8 only flip bits [3:0] -> partner stays in the same half.
    #pragma unroll
    for (int m = 1; m <= 8; m <<= 1) {
        #pragma unroll
        for (int r = 0; r < 8; ++r) {
            unsigned b0 = (unsigned)__shfl_xor((int)t0[r], m, 32);
            unsigned b1 = (unsigned)__shfl_xor((int)t1[r], m, 32);
            unsigned b2 = (unsigned)__shfl_xor((int)t2[r], m, 32);
            ins3u(b0, t0[r], t1[r], t2[r]);
            ins3u(b1, t0[r], t1[r], t2[r]);
            ins3u(b2, t0[r], t1[r], t2[r]);
        }
    }

    // Lane l==0 of each half deposits its wave's merged triples into LDS.
    if (l == 0) {
        #pragma unroll
        for (int r = 0; r < 8; ++r) {
            red[wave][r + 8 * half][0] = t0[r];
            red[wave][r + 8 * half][1] = t1[r];
            red[wave][r + 8 * half][2] = t2[r];
        }
    }
    __syncthreads();

    // Threads 0-15: final 4-way merge for one row each, then scale.
    if (tid < 16) {
        unsigned m0 = red[0][tid][0];
        unsigned m1 = red[0][tid][1];
        unsigned m2 = red[0][tid][2];
        #pragma unroll
        for (int w = 1; w < 4; ++w) {
            ins3u(red[w][tid][0], m0, m1, m2);
            ins3u(red[w][tid][1], m0, m1, m2);
            ins3u(red[w][tid][2], m0, m1, m2);
        }
        float mean = (sqrtf(__uint_as_float(m0)) +
                      sqrtf(__uint_as_float(m1)) +
                      sqrtf(__uint_as_float(m2))) * (1.0f / 3.0f);
        float ms = fmaxf(mean, 1e-5f);
        out9[(size_t)(rowBase + tid) * 9] = BASE_SCALE * ms;
    }
}

// Per point: normalize quaternion, build rotation, cov = s^2 * R * R^T.
__global__ __launch_bounds__(256) void cov_kernel(const float* __restrict__ quat,
                                                  float* __restrict__ out9) {
    int i = blockIdx.x * blockDim.x + threadIdx.x;
    if (i >= N_PTS) return;

    float qr = quat[i * 4 + 0];
    float qx = quat[i * 4 + 1];
    float qy = quat[i * 4 + 2];
    float qz = quat[i * 4 + 3];
    float inv = 1.0f / sqrtf(qr * qr + qx * qx + qy * qy + qz * qz);
    qr *= inv; qx *= inv; qy *= inv; qz *= inv;

    float R[3][3];
    R[0][0] = 1.0f - 2.0f * (qy * qy + qz * qz);
    R[0][1] = 2.0f * (qx * qy - qr * qz);
    R[0][2] = 2.0f * (qx * qz + qr * qy);
    R[1][0] = 2.0f * (qx * qy + qr * qz);
    R[1][1] = 1.0f - 2.0f * (qx * qx + qz * qz);
    R[1][2] = 2.0f * (qy * qz - qr * qx);
    R[2][0] = 2.0f * (qx * qz - qr * qy);
    R[2][1] = 2.0f * (qy * qz + qr * qx);
    R[2][2] = 1.0f - 2.0f * (qx * qx + qy * qy);

    float s = out9[(size_t)i * 9];   // stashed by knn_scale_kernel
    float s2 = s * s;

    float* o = out9 + (size_t)i * 9;
    #pragma unroll
    for (int aa = 0; aa < 3; ++aa) {
        #pragma unroll
        for (int bb = 0; bb < 3; ++bb) {
            float v = R[aa][0] * R[bb][0] + R[aa][1] * R[bb][1] + R[aa][2] * R[bb][2];
            o[aa * 3 + bb] = s2 * v;
        }
    }
}

extern "C" void kernel_launch(void* const* d_in, const int* in_sizes, int n_in,
                              void* d_out, int out_size, void* d_ws, size_t ws_size,
                              hipStream_t stream) {
    const float* pts  = (const float*)d_in[0];   // (N, 3) float32
    const float* quat = (const float*)d_in[1];   // (N, 4) float32
    float* out = (float*)d_out;                  // (N, 3, 3) float32

    (void)in_sizes; (void)n_in; (void)d_ws; (void)ws_size; (void)out_size;

    // 768 row tiles of 16 points; 4 waves per tile splitting the column range.
    knn_scale_kernel<<<N_PTS / 16, 128, 0, stream>>>(pts, out);
    cov_kernel<<<(N_PTS + 255) / 256, 256, 0, stream>>>(quat, out);
}